// GRU_Gen_78262894068261
// MI455X (gfx1250) — compile-verified
//
#include <hip/hip_runtime.h>
#include <hip/hip_bf16.h>

// Problem constants (from the reference)
#define TSTEPS 1024
#define BB 512
#define HH 15
#define II 10
#define OO 128

typedef __attribute__((ext_vector_type(2))) float v2f;
typedef __attribute__((ext_vector_type(4))) float v4f;
typedef __attribute__((ext_vector_type(8))) float v8f;

// D = A(16x4 f32) x B(4x16 f32) + C(16x16 f32)
__device__ __forceinline__ v8f wmma_f32_16x16x4(v2f a, v2f b, v8f c) {
  return __builtin_amdgcn_wmma_f32_16x16x4_f32(
      /*neg_a=*/false, a, /*neg_b=*/false, b,
      /*c_mod=*/(short)0, c, /*reuse_a=*/false, /*reuse_b=*/false);
}

// Swap values between lane L and lane L^16 (half-wave swap) via ds_bpermute.
__device__ __forceinline__ float lane_swap16(float v) {
  return __int_as_float(__builtin_amdgcn_ds_bpermute(
      (int)((((int)threadIdx.x & 31) ^ 16) << 2), __float_as_int(v)));
}

// v_rcp_f32-based fast nonlinearities (~1 ulp rcp, avoids IEEE divide sequence)
__device__ __forceinline__ float fast_sigmoid(float x) {
  return __builtin_amdgcn_rcpf(1.0f + __expf(-x));
}
__device__ __forceinline__ float fast_tanh(float x) {
  // tanh(x) = 1 - 2/(exp(2x)+1)
  return 1.0f - 2.0f * __builtin_amdgcn_rcpf(__expf(2.0f * x) + 1.0f);
}

// ---------------------------------------------------------------------------
// Kernel 1: sequential GRU scan. One wave (32 threads) per 16-batch tile.
//   A (registers, invariant): Wpad = W_hh padded per-gate to 3 tiles of 16x16
//   B (rebuilt each step):    h^T as 16(K=j) x 16(N=batch)
//   r/z gates: b_hh folded into Gi, WMMA chain starts from C = 0
//   n gate:    b_hh kept as C tile (it is scaled by r inside tanh)
//   D layout: VGPR v -> row (v | v+8) across lane halves, col = lane%16 = batch
// Stores h_t into hs (T, B, 16) fp32 (padded H -> aligned b128 stores).
// ---------------------------------------------------------------------------
__global__ __launch_bounds__(32)
void gru_scan_kernel(const float* __restrict__ feed, const float* __restrict__ h0,
                     const float* __restrict__ W_ih, const float* __restrict__ W_hh,
                     const float* __restrict__ b_ih, const float* __restrict__ b_hh,
                     float* __restrict__ hs) {
  const int lane = threadIdx.x;
  const int col  = lane & 15;          // batch column within tile / M row for A
  const bool hi  = lane >= 16;
  const int b    = (blockIdx.x << 4) + col;   // global batch element
  const int jofs = hi ? 8 : 0;         // D-layout row offset of this half-wave

  // ---- A tiles: Wpad[g*16+M][K], A-layout: lanes0-15 K=k0..k0+1, lanes16-31 K=k0+2..k0+3
  v2f A[3][4];
#pragma unroll
  for (int g = 0; g < 3; ++g) {
#pragma unroll
    for (int s = 0; s < 4; ++s) {
      const int m = col;
      const int k0 = 4 * s + (hi ? 2 : 0);
      float a0 = 0.0f, a1 = 0.0f;
      if (m < HH) {
        const float* wr = W_hh + (g * HH + m) * HH;
        if (k0 < HH)     a0 = wr[k0];
        if (k0 + 1 < HH) a1 = wr[k0 + 1];
      }
      A[g][s].x = a0;
      A[g][s].y = a1;
    }
  }

  // ---- C tile for the n gate only: b_hh_n broadcast in D layout
  v8f Cbn;
#pragma unroll
  for (int v = 0; v < 8; ++v) {
    const int j = v + jofs;
    Cbn[v] = (j < HH) ? b_hh[2 * HH + j] : 0.0f;
  }

  // ---- gi = x @ W_ih^T + b_ih (+ b_hh for r/z gates), once, into D-layout tiles
  float x[II];
#pragma unroll
  for (int i = 0; i < II; ++i) x[i] = feed[b * II + i];

  v8f Gi[3];
#pragma unroll
  for (int g = 0; g < 3; ++g) {
#pragma unroll
    for (int v = 0; v < 8; ++v) {
      const int j = v + jofs;
      float acc = 0.0f;
      if (j < HH) {
        const float* wr = W_ih + (g * HH + j) * II;
        acc = b_ih[g * HH + j];
        if (g < 2) acc += b_hh[g * HH + j];   // fold additive b_hh for r,z
#pragma unroll
        for (int i = 0; i < II; ++i) acc = fmaf(x[i], wr[i], acc);
      }
      Gi[g][v] = acc;
    }
  }

  // ---- h0 in D layout (pad row 15 = 0)
  v8f h;
#pragma unroll
  for (int v = 0; v < 8; ++v) {
    const int j = v + jofs;
    h[v] = (j < HH) ? h0[b * HH + j] : 0.0f;
  }

  const v8f czero = { 0.0f, 0.0f, 0.0f, 0.0f, 0.0f, 0.0f, 0.0f, 0.0f };
  float* dst = hs + (((size_t)b) << 4) + jofs;
  const size_t tstride = (size_t)BB * 16;

  for (int t = 0; t < TSTEPS; ++t) {
    // ---- Relayout h from C/D layout to B layout (4 K-slices of 4x16).
    // B slice s VGPR0: lanes0-15 K=4s, lanes16-31 K=4s+2; VGPR1: K=4s+1 / 4s+3.
    // h rows 0-7 live in lanes0-15, rows 8-15 in lanes16-31 -> half-wave swaps.
    const float sw0 = lane_swap16(h[0]), sw1 = lane_swap16(h[1]);
    const float sw2 = lane_swap16(h[2]), sw3 = lane_swap16(h[3]);
    const float sw4 = lane_swap16(h[4]), sw5 = lane_swap16(h[5]);
    const float sw6 = lane_swap16(h[6]), sw7 = lane_swap16(h[7]);
    v2f Bm[4];
    Bm[0].x = hi ? sw2 : h[0];  Bm[0].y = hi ? sw3 : h[1];   // K = 0,2 / 1,3
    Bm[1].x = hi ? sw6 : h[4];  Bm[1].y = hi ? sw7 : h[5];   // K = 4,6 / 5,7
    Bm[2].x = hi ? h[2] : sw0;  Bm[2].y = hi ? h[3] : sw1;   // K = 8,10 / 9,11
    Bm[3].x = hi ? h[6] : sw4;  Bm[3].y = hi ? h[7] : sw5;   // K = 12,14 / 13,15

    // ---- gate pre-activations: 3 independent 4-deep WMMA chains
    v8f D0 = czero, D1 = czero, D2 = Cbn;
#pragma unroll
    for (int s = 0; s < 4; ++s) {
      D0 = wmma_f32_16x16x4(A[0][s], Bm[s], D0);
      D1 = wmma_f32_16x16x4(A[1][s], Bm[s], D1);
      D2 = wmma_f32_16x16x4(A[2][s], Bm[s], D2);
    }

    // ---- pointwise GRU update, all in D layout
#pragma unroll
    for (int v = 0; v < 8; ++v) {
      const float r = fast_sigmoid(Gi[0][v] + D0[v]);
      const float z = fast_sigmoid(Gi[1][v] + D1[v]);
      const float n = fast_tanh(Gi[2][v] + r * D2[v]);
      h[v] = fmaf(z, h[v] - n, n);   // (1-z)*n + z*h
    }

    // ---- store h_t: lane<16 writes j=0..7, lane>=16 writes j=8..15 (15 = pad)
    v4f lo4 = { h[0], h[1], h[2], h[3] };
    v4f hi4 = { h[4], h[5], h[6], h[7] };
    *(v4f*)(dst)     = lo4;
    *(v4f*)(dst + 4) = hi4;
    dst += tstride;
  }
}

// ---------------------------------------------------------------------------
// Kernel 2: out = sigmoid(hs @ W_out^T + b_out). Each wave handles 4
// consecutive 16x128 tiles, amortizing the invariant W_out B-tile preload.
// hs (32 MB) stays resident in the 192 MB L2; output stores are nontemporal.
// ---------------------------------------------------------------------------
#define TILES_PER_WAVE 4
__global__ __launch_bounds__(256)
void gru_out_kernel(const float* __restrict__ hs, const float* __restrict__ W_out,
                    const float* __restrict__ b_out, float* __restrict__ out) {
  const int lane = (int)threadIdx.x & 31;
  const int col  = lane & 15;
  const bool hi  = lane >= 16;
  const int wave = ((int)blockIdx.x << 3) + ((int)threadIdx.x >> 5);

  // ---- B tiles: W_out^T (K=j 0..15, N=o). Row 15 padded with zeros. Invariant.
  v2f Bw[8][4];
  float bias[8];
#pragma unroll
  for (int nt = 0; nt < 8; ++nt) {
    const int o = (nt << 4) + col;
    const float* wr = W_out + o * HH;
    bias[nt] = b_out[o];
#pragma unroll
    for (int s = 0; s < 4; ++s) {
      const int k0 = 4 * s + (hi ? 2 : 0);
      Bw[nt][s].x = (k0 < HH)     ? wr[k0]     : 0.0f;
      Bw[nt][s].y = (k0 + 1 < HH) ? wr[k0 + 1] : 0.0f;
    }
  }

#pragma unroll 1
  for (int it = 0; it < TILES_PER_WAVE; ++it) {
    const int tile = wave * TILES_PER_WAVE + it;   // 0..32767
    const int t    = tile >> 5;                    // 32 batch-tiles per timestep
    const int b0   = (tile & 31) << 4;

    // ---- A tile: 16 padded hs rows (64B contiguous each), K-quadrant select
    const float* arow = hs + (((size_t)t * BB + b0 + col) << 4);
    const v4f r0 = *(const v4f*)(arow);
    const v4f r1 = *(const v4f*)(arow + 4);
    const v4f r2 = *(const v4f*)(arow + 8);
    const v4f r3 = *(const v4f*)(arow + 12);
    v2f Am[4];
    Am[0].x = hi ? r0.z : r0.x;  Am[0].y = hi ? r0.w : r0.y;
    Am[1].x = hi ? r1.z : r1.x;  Am[1].y = hi ? r1.w : r1.y;
    Am[2].x = hi ? r2.z : r2.x;  Am[2].y = hi ? r2.w : r2.y;
    Am[3].x = hi ? r3.z : r3.x;  Am[3].y = hi ? r3.w : r3.y;

    const size_t outRowBase = ((size_t)t * BB + b0 + (hi ? 8 : 0)) * OO;
#pragma unroll
    for (int nt = 0; nt < 8; ++nt) {
      const float bs = bias[nt];
      v8f acc = { bs, bs, bs, bs, bs, bs, bs, bs };
#pragma unroll
      for (int s = 0; s < 4; ++s)
        acc = wmma_f32_16x16x4(Am[s], Bw[nt][s], acc);
#pragma unroll
      for (int v = 0; v < 8; ++v) {
        const float val = fast_sigmoid(acc[v]);
        __builtin_nontemporal_store(
            val, out + outRowBase + (size_t)v * OO + (nt << 4) + col);
      }
    }
  }
}

extern "C" void kernel_launch(void* const* d_in, const int* in_sizes, int n_in,
                              void* d_out, int out_size, void* d_ws, size_t ws_size,
                              hipStream_t stream) {
  (void)in_sizes; (void)n_in; (void)out_size; (void)ws_size;
  const float* feed  = (const float*)d_in[0];
  const float* h0    = (const float*)d_in[1];
  const float* W_ih  = (const float*)d_in[2];
  const float* W_hh  = (const float*)d_in[3];
  const float* b_ih  = (const float*)d_in[4];
  const float* b_hh  = (const float*)d_in[5];
  const float* W_out = (const float*)d_in[6];
  const float* b_out = (const float*)d_in[7];
  float* out = (float*)d_out;
  float* hs  = (float*)d_ws;  // (T, B, 16) fp32 = 32 MiB scratch, fits in L2

  // 32 independent 16-batch tiles, one wave each (recurrence is latency-bound).
  gru_scan_kernel<<<BB / 16, 32, 0, stream>>>(feed, h0, W_ih, W_hh, b_ih, b_hh, hs);
  // 32768 output tiles of 16x128; 4 tiles per wave, 8 waves per block.
  gru_out_kernel<<<(TSTEPS * BB / 16) / (8 * TILES_PER_WAVE), 256, 0, stream>>>(
      hs, W_out, b_out, out);
}